// PointNet2RegMSG_75273596829890
// MI455X (gfx1250) — compile-verified
//
#include <hip/hip_runtime.h>
#include <cstddef>
#include <cstdint>

typedef float v2f __attribute__((ext_vector_type(2)));
typedef float v8f __attribute__((ext_vector_type(8)));

#define GEMM_WAVES 8

// ---------------------------------------------------------------- transpose
// xyz (B,6,N) -> coords (B,N,3), norms (B,N,3)
__global__ void k_transpose(const float* __restrict__ x, float* __restrict__ coords,
                            float* __restrict__ norms, int N, int total) {
  int i = blockIdx.x * blockDim.x + threadIdx.x;
  if (i >= total) return;                       // total = B*N
  int b = i / N, n = i % N;
  const float* src = x + (size_t)b * 6 * N;
  float* c = coords + (size_t)i * 3;
  float* m = norms  + (size_t)i * 3;
  c[0] = src[0 * N + n]; c[1] = src[1 * N + n]; c[2] = src[2 * N + n];
  m[0] = src[3 * N + n]; m[1] = src[4 * N + n]; m[2] = src[5 * N + n];
}

// ---------------------------------------------------------------- FPS
__global__ void k_fps(const float* __restrict__ xyz, int N, int S, int* __restrict__ idx) {
  __shared__ float dist[4096];
  __shared__ float rmax[256];
  __shared__ int   rarg[256];
  __shared__ int   cur;
  int b = blockIdx.x;
  int t = threadIdx.x;
  const float* p = xyz + (size_t)b * N * 3;
  for (int i = t; i < N; i += 256) dist[i] = 1e10f;
  if (t == 0) cur = 0;
  __syncthreads();
  for (int s = 0; s < S; s++) {
    int far = cur;
    if (t == 0) idx[b * S + s] = far;
    float cx = p[far * 3 + 0], cy = p[far * 3 + 1], cz = p[far * 3 + 2];
    float best = -1.0f; int barg = 0;
    for (int i = t; i < N; i += 256) {
      float dx = p[i * 3 + 0] - cx;
      float dy = p[i * 3 + 1] - cy;
      float dz = p[i * 3 + 2] - cz;
      float d = dx * dx + dy * dy + dz * dz;
      float nd = fminf(dist[i], d);
      dist[i] = nd;
      if (nd > best) { best = nd; barg = i; }
    }
    rmax[t] = best; rarg[t] = barg;
    __syncthreads();
    for (int o = 128; o > 0; o >>= 1) {
      if (t < o) {
        if (rmax[t + o] > rmax[t] ||
            (rmax[t + o] == rmax[t] && rarg[t + o] < rarg[t])) {
          rmax[t] = rmax[t + o]; rarg[t] = rarg[t + o];
        }
      }
      __syncthreads();
    }
    if (t == 0) cur = rarg[0];
    __syncthreads();
  }
}

// ---------------------------------------------------------------- gather xyz by index
__global__ void k_gather3(const float* __restrict__ src, const int* __restrict__ idx,
                          float* __restrict__ dst, int N, int S, int total) {
  int i = blockIdx.x * blockDim.x + threadIdx.x;
  if (i >= total) return;                       // total = B*S
  int b = i / S;
  int g = idx[i];
  const float* s = src + ((size_t)b * N + g) * 3;
  float* d = dst + (size_t)i * 3;
  d[0] = s[0]; d[1] = s[1]; d[2] = s[2];
}

// ---------------------------------------------------------------- ball query
__global__ void k_ballquery(const float* __restrict__ xyz, int N,
                            const float* __restrict__ nxyz, int S,
                            float r2, int K, int* __restrict__ gi, int total) {
  int q = blockIdx.x * blockDim.x + threadIdx.x;
  if (q >= total) return;                       // total = B*S
  int b = q / S;
  const float* P = xyz + (size_t)b * N * 3;
  float ax = nxyz[q * 3 + 0], ay = nxyz[q * 3 + 1], az = nxyz[q * 3 + 2];
  float na = ax * ax + ay * ay + az * az;
  int* out = gi + (size_t)q * K;
  int cnt = 0, first = 0;
  for (int i = 0; i < N && cnt < K; i++) {
    float bx = P[i * 3 + 0], by = P[i * 3 + 1], bz = P[i * 3 + 2];
    float d = na - 2.0f * (ax * bx + ay * by + az * bz) + (bx * bx + by * by + bz * bz);
    if (!(d > r2)) {
      if (cnt == 0) first = i;
      out[cnt++] = i;
    }
  }
  for (; cnt < K; cnt++) out[cnt] = first;
}

// ---------------------------------------------------------------- grouped feature build
__global__ void k_build_feat(const float* __restrict__ pts, int Cp, int ldp,
                             const float* __restrict__ xyz, const float* __restrict__ nxyz,
                             const int* __restrict__ gi, int N, int S, int K,
                             int Cpad, float* __restrict__ feat, int total) {
  int row = blockIdx.x * blockDim.x + threadIdx.x;
  if (row >= total) return;                     // total = B*S*K
  int q = row / K;
  int b = q / S;
  int g = gi[row];
  float* f = feat + (size_t)row * Cpad;
  const float* src = pts + ((size_t)b * N + g) * ldp;
  for (int c = 0; c < Cp; c++) f[c] = src[c];
  const float* xg = xyz + ((size_t)b * N + g) * 3;
  f[Cp + 0] = xg[0] - nxyz[q * 3 + 0];
  f[Cp + 1] = xg[1] - nxyz[q * 3 + 1];
  f[Cp + 2] = xg[2] - nxyz[q * 3 + 2];
  for (int c = Cp + 3; c < Cpad; c++) f[c] = 0.0f;
}

__global__ void k_feat_all(const float* __restrict__ pts, int Cp,
                           const float* __restrict__ xyz, int Cpad,
                           float* __restrict__ feat, int total) {
  int i = blockIdx.x * blockDim.x + threadIdx.x;
  if (i >= total) return;
  float* f = feat + (size_t)i * Cpad;
  const float* s = pts + (size_t)i * Cp;
  for (int c = 0; c < Cp; c++) f[c] = s[c];
  f[Cp + 0] = xyz[i * 3 + 0];
  f[Cp + 1] = xyz[i * 3 + 1];
  f[Cp + 2] = xyz[i * 3 + 2];
  for (int c = Cp + 3; c < Cpad; c++) f[c] = 0.0f;
}

// ---------------------------------------------------------------- zero
__global__ void k_zero(float* __restrict__ p, int n) {
  int i = blockIdx.x * blockDim.x + threadIdx.x;
  if (i < n) p[i] = 0.0f;
}

// ---------------------------------------------------------------- weight/bias pad + interleave
// Wp layout: Wp[(k/4)*(Opad*4) + col*4 + (k%4)] — one k-step's B fragments for a
// whole 128-col block are contiguous (2KB), so GEMM B loads use immediate offsets.
__global__ void k_padW(const float* __restrict__ W, int Opad, int Oact, int C, int ldk,
                       float* __restrict__ Wp, const float* __restrict__ b,
                       float* __restrict__ bp, int total) {
  int i = blockIdx.x * blockDim.x + threadIdx.x;
  if (i < Opad) bp[i] = (i < Oact) ? b[i] : 0.0f;
  if (i >= total) return;                       // total = Opad*ldk
  int col = i / ldk, k = i % ldk;
  float v = (col < Oact && k < C) ? W[(size_t)col * C + k] : 0.0f;
  Wp[(size_t)(k >> 2) * (Opad * 4) + col * 4 + (k & 3)] = v;
}

// ---------------------------------------------------------------- WMMA GEMM
// Y[R,O] = act(X[R,ldk]) @ W^T + bias, W pre-padded & k-interleaved.
// One wave computes a 16x(16*NCOL) tile; per 4-k step: 1 A b64 load, NCOL
// batched B b64 loads off one pointer (imm offsets), NCOL WMMAs.
template <int NCOL, bool HASBN>
__global__ void k_gemm_wmma(const float* __restrict__ X, int ldk,
                            const float* __restrict__ Wp, const float* __restrict__ bp,
                            float* __restrict__ Y, int R, int O,
                            const float* __restrict__ inSc, const float* __restrict__ inSh,
                            float* __restrict__ sumOut, float* __restrict__ sumsqOut) {
  __shared__ float lsum[16 * NCOL];
  __shared__ float lss[16 * NCOL];
  __shared__ float sBn[2048];
  const int tid  = threadIdx.x;
  const int lane = tid & 31;
  const int wave = tid >> 5;
  const int n    = lane & 15;   // A-row (M) / B-column (N) lane index
  const int half = lane >> 4;   // K sub-pair / M upper half
  const int colBase = blockIdx.y * (16 * NCOL);
  const int rowTile = blockIdx.x * GEMM_WAVES + wave;
  if (tid < 16 * NCOL) { lsum[tid] = 0.0f; lss[tid] = 0.0f; }
  if constexpr (HASBN) {
    for (int i = tid; i < ldk; i += 256) { sBn[i] = inSc[i]; sBn[1024 + i] = inSh[i]; }
  }
  __syncthreads();
  float psum[NCOL], pss[NCOL];
#pragma unroll
  for (int j = 0; j < NCOL; j++) { psum[j] = 0.0f; pss[j] = 0.0f; }
  if (rowTile * 16 < R) {
    const int row0 = rowTile * 16;
    const float* xrow = X + (size_t)(row0 + n) * ldk + 2 * half;
    // B pointer: k-block base + lane offset (col*4 + 2*half); tile j at +64 floats
    const float* wptr = Wp + (size_t)colBase * 4 + n * 4 + 2 * half;
    const size_t wstep = (size_t)O * 4;
    v8f acc[NCOL];
#pragma unroll
    for (int j = 0; j < NCOL; j++) {
      const float bv = bp[colBase + 16 * j + n];
#pragma unroll
      for (int v = 0; v < 8; v++) acc[j][v] = bv;
    }
    for (int k0 = 0; k0 < ldk; k0 += 4) {
      float a0 = xrow[k0];
      float a1 = xrow[k0 + 1];
      if constexpr (HASBN) {
        const int ka = k0 + 2 * half;
        a0 = fmaxf(fmaf(a0, sBn[ka],     sBn[1024 + ka]),     0.0f);
        a1 = fmaxf(fmaf(a1, sBn[ka + 1], sBn[1024 + ka + 1]), 0.0f);
      }
      v2f av; av[0] = a0; av[1] = a1;
      v2f bw[NCOL];
#pragma unroll
      for (int j = 0; j < NCOL; j++) {
        bw[j] = *(const v2f*)(wptr + 64 * j);   // imm offset 256B * j
      }
#pragma unroll
      for (int j = 0; j < NCOL; j++) {
        acc[j] = __builtin_amdgcn_wmma_f32_16x16x4_f32(false, av, false, bw[j],
                                                       (short)0, acc[j], false, false);
      }
      wptr += wstep;
    }
#pragma unroll
    for (int j = 0; j < NCOL; j++) {
      float* ybase = Y + (size_t)row0 * O + colBase + 16 * j;
#pragma unroll
      for (int v = 0; v < 8; v++) {
        const int m = v + 8 * half;             // C/D layout: VGPR v + 8*half = row M
        float val = acc[j][v];
        ybase[(size_t)m * O + n] = val;
        psum[j] += val; pss[j] += val * val;
      }
    }
  }
  if (sumOut) {
#pragma unroll
    for (int j = 0; j < NCOL; j++) {
      atomicAdd(&lsum[16 * j + n], psum[j]);
      atomicAdd(&lss[16 * j + n],  pss[j]);
    }
    __syncthreads();
    if (tid < 16 * NCOL) {
      atomicAdd(&sumOut[colBase + tid],   lsum[tid]);
      atomicAdd(&sumsqOut[colBase + tid], lss[tid]);
    }
  }
}

// ---------------------------------------------------------------- BN finalize
__global__ void k_finalize_bn(const float* __restrict__ sum, const float* __restrict__ sumsq,
                              float cnt, const float* __restrict__ g, const float* __restrict__ be,
                              float* __restrict__ scale, float* __restrict__ shift, int O) {
  int o = blockIdx.x * blockDim.x + threadIdx.x;
  if (o >= O) return;
  float mu  = sum[o] / cnt;
  float var = fmaxf(sumsq[o] / cnt - mu * mu, 0.0f);
  float sc  = g[o] * rsqrtf(var + 1e-5f);
  scale[o] = sc;
  shift[o] = be[o] - mu * sc;
}

// ---------------------------------------------------------------- BN+ReLU + max over K
__global__ void k_maxpool(const float* __restrict__ Y, int K, int O,
                          const float* __restrict__ scale, const float* __restrict__ shift,
                          float* __restrict__ out, int ldo, int coff, int total) {
  int i = blockIdx.x * blockDim.x + threadIdx.x;
  if (i >= total) return;                       // total = B*S*O
  int o = i % O, q = i / O;
  const float* y = Y + ((size_t)q * K) * O + o;
  float sc = scale[o], sh = shift[o];
  float m = -1e30f;
  for (int k = 0; k < K; k++) {
    float v = fmaxf(fmaf(y[(size_t)k * O], sc, sh), 0.0f);
    m = fmaxf(m, v);
  }
  out[(size_t)q * ldo + coff + o] = m;
}

// ---------------------------------------------------------------- final copy (strip O-pad)
__global__ void k_copyout(const float* __restrict__ Y, float* __restrict__ out, int total) {
  int i = blockIdx.x * blockDim.x + threadIdx.x;
  if (i >= total) return;                       // total = 16*27, Y is 16x32
  out[i] = Y[(i / 27) * 32 + (i % 27)];
}

// ================================================================ host side
struct LayerP { const float* W; const float* b; const float* g; const float* be; int cin; int cout; };

static void launch_gemm(hipStream_t st, const float* X, int ldk,
                        const float* Wp, const float* bp,
                        float* Y, int R, int O,
                        const float* sc, const float* sh,
                        float* sum, float* sumsq) {
  int tiles = O / 16;
  int ncol = (tiles % 8 == 0) ? 8 : (tiles % 4 == 0) ? 4 : (tiles % 2 == 0) ? 2 : 1;
  dim3 grid((unsigned)((R / 16 + GEMM_WAVES - 1) / GEMM_WAVES), (unsigned)(tiles / ncol));
  bool hb = (sc != nullptr);
#define GO(NC)                                                                              \
  do {                                                                                      \
    if (hb) k_gemm_wmma<NC, true><<<grid, dim3(256), 0, st>>>(X, ldk, Wp, bp, Y, R, O,      \
                                                              sc, sh, sum, sumsq);          \
    else    k_gemm_wmma<NC, false><<<grid, dim3(256), 0, st>>>(X, ldk, Wp, bp, Y, R, O,     \
                                                               sc, sh, sum, sumsq);         \
  } while (0)
  switch (ncol) {
    case 8: GO(8); break;
    case 4: GO(4); break;
    case 2: GO(2); break;
    default: GO(1); break;
  }
#undef GO
}

static void run_mlp(hipStream_t stream, const LayerP* L, int nL,
                    float* bufA, float* bufB, int R, int Cpad0,
                    float* stats, float* wpad, float* bpad,
                    float* scale0, float* shift0, float* scale1, float* shift1,
                    float** outY, const float** outScale, const float** outShift) {
  const float* X = bufA; int ldk = Cpad0;
  const float* inSc = nullptr; const float* inSh = nullptr;
  float* Y = nullptr;
  float* scl[2] = { scale0, scale1 };
  float* sft[2] = { shift0, shift1 };
  for (int l = 0; l < nL; l++) {
    Y = (l & 1) ? bufA : bufB;
    int O = L[l].cout;
    int padTotal = O * ldk;
    k_padW<<<dim3((padTotal + 255) / 256), dim3(256), 0, stream>>>(
        L[l].W, O, O, L[l].cin, ldk, wpad, L[l].b, bpad, padTotal);
    k_zero<<<dim3(8), dim3(256), 0, stream>>>(stats, 2048);
    launch_gemm(stream, X, ldk, wpad, bpad, Y, R, O, inSc, inSh, stats, stats + 1024);
    int slot = l & 1;
    k_finalize_bn<<<dim3((O + 255) / 256), dim3(256), 0, stream>>>(
        stats, stats + 1024, (float)R, L[l].g, L[l].be, scl[slot], sft[slot], O);
    X = Y; ldk = O; inSc = scl[slot]; inSh = sft[slot];
  }
  *outY = Y; *outScale = inSc; *outShift = inSh;
}

extern "C" void kernel_launch(void* const* d_in, const int* in_sizes, int n_in,
                              void* d_out, int out_size, void* d_ws, size_t ws_size,
                              hipStream_t stream) {
  (void)in_sizes; (void)n_in; (void)out_size; (void)ws_size;
  const int Bn = 16, N = 4096, S1 = 512, S2 = 128;

  // -------- unpack params (pytree flatten order of setup_inputs)
  int p = 0;
  const float* xin = (const float*)d_in[p++];
  LayerP sa1[3][3], sa2[3][3], sa3[3], fcl[2];
  const int sa1ch[3][3] = { {32, 32, 64}, {64, 64, 128}, {64, 96, 128} };
  const int sa2ch[3][3] = { {64, 64, 128}, {128, 128, 256}, {128, 128, 256} };
  const int sa3ch[3] = { 256, 512, 1024 };
  for (int br = 0; br < 3; br++) { int cin = 6;
    for (int l = 0; l < 3; l++) {
      sa1[br][l] = { (const float*)d_in[p], (const float*)d_in[p + 1],
                     (const float*)d_in[p + 2], (const float*)d_in[p + 3], cin, sa1ch[br][l] };
      p += 4; cin = sa1ch[br][l];
    } }
  for (int br = 0; br < 3; br++) { int cin = 323;
    for (int l = 0; l < 3; l++) {
      sa2[br][l] = { (const float*)d_in[p], (const float*)d_in[p + 1],
                     (const float*)d_in[p + 2], (const float*)d_in[p + 3], cin, sa2ch[br][l] };
      p += 4; cin = sa2ch[br][l];
    } }
  { int cin = 643;
    for (int l = 0; l < 3; l++) {
      sa3[l] = { (const float*)d_in[p], (const float*)d_in[p + 1],
                 (const float*)d_in[p + 2], (const float*)d_in[p + 3], cin, sa3ch[l] };
      p += 4; cin = sa3ch[l];
    } }
  fcl[0] = { (const float*)d_in[p], (const float*)d_in[p + 1],
             (const float*)d_in[p + 2], (const float*)d_in[p + 3], 1024, 512 }; p += 4;
  fcl[1] = { (const float*)d_in[p], (const float*)d_in[p + 1],
             (const float*)d_in[p + 2], (const float*)d_in[p + 3], 512, 256 }; p += 4;
  const float* W3 = (const float*)d_in[p++];
  const float* b3 = (const float*)d_in[p++];

  // -------- workspace layout
  uint8_t* base = (uint8_t*)d_ws; size_t off = 0;
  auto alloc = [&](size_t nbytes) -> void* {
    void* r = base + off; off = (off + nbytes + 255) & ~(size_t)255; return r;
  };
  float* coords = (float*)alloc((size_t)Bn * N * 3 * 4);
  float* norms  = (float*)alloc((size_t)Bn * N * 3 * 4);
  float* l1x    = (float*)alloc((size_t)Bn * S1 * 3 * 4);
  float* l1p    = (float*)alloc((size_t)Bn * S1 * 320 * 4);
  float* l2x    = (float*)alloc((size_t)Bn * S2 * 3 * 4);
  float* l2p    = (float*)alloc((size_t)Bn * S2 * 640 * 4);
  float* gbuf   = (float*)alloc((size_t)Bn * 1024 * 4);
  int*   fpsidx = (int*)alloc((size_t)Bn * S1 * 4);
  int*   gi     = (int*)alloc((size_t)Bn * S1 * 128 * 4);
  float* stats  = (float*)alloc(2048 * 4);
  float* scale0 = (float*)alloc(1024 * 4);
  float* shift0 = (float*)alloc(1024 * 4);
  float* scale1 = (float*)alloc(1024 * 4);
  float* shift1 = (float*)alloc(1024 * 4);
  float* wpad   = (float*)alloc((size_t)1048576 * 4);    // max padded weight 1024x1024
  float* bpad   = (float*)alloc(1024 * 4);
  float* bufA   = (float*)alloc((size_t)100663296 * 4);  // 16*512*128*96
  float* bufB   = (float*)alloc((size_t)134217728 * 4);  // 16*512*128*128

  const int TPB = 256;
  auto blk = [&](long n) { return dim3((unsigned)((n + TPB - 1) / TPB)); };

  // -------- stage 0: split input
  k_transpose<<<blk((long)Bn * N), dim3(TPB), 0, stream>>>(xin, coords, norms, N, Bn * N);

  // -------- SA1 (MSG on 4096 pts -> 512 centroids)
  k_fps<<<dim3(Bn), dim3(256), 0, stream>>>(coords, N, S1, fpsidx);
  k_gather3<<<blk((long)Bn * S1), dim3(TPB), 0, stream>>>(coords, fpsidx, l1x, N, S1, Bn * S1);
  {
    const float rads[3] = { 0.1f, 0.2f, 0.4f };
    const int   Ks[3]   = { 16, 32, 128 };
    const int   coff[3] = { 0, 64, 192 };
    for (int br = 0; br < 3; br++) {
      int K = Ks[br]; int R = Bn * S1 * K;
      float r2 = rads[br] * rads[br];
      k_ballquery<<<blk((long)Bn * S1), dim3(TPB), 0, stream>>>(coords, N, l1x, S1, r2, K, gi, Bn * S1);
      k_build_feat<<<blk((long)R), dim3(TPB), 0, stream>>>(norms, 3, 3, coords, l1x, gi,
                                                           N, S1, K, 8, bufA, R);
      float* Y; const float* sc; const float* sh;
      run_mlp(stream, sa1[br], 3, bufA, bufB, R, 8, stats, wpad, bpad,
              scale0, shift0, scale1, shift1, &Y, &sc, &sh);
      int O = sa1[br][2].cout;
      k_maxpool<<<blk((long)Bn * S1 * O), dim3(TPB), 0, stream>>>(Y, K, O, sc, sh,
                                                                  l1p, 320, coff[br], Bn * S1 * O);
    }
  }

  // -------- SA2 (MSG on 512 pts -> 128 centroids)
  k_fps<<<dim3(Bn), dim3(256), 0, stream>>>(l1x, S1, S2, fpsidx);
  k_gather3<<<blk((long)Bn * S2), dim3(TPB), 0, stream>>>(l1x, fpsidx, l2x, S1, S2, Bn * S2);
  {
    const float rads[3] = { 0.2f, 0.4f, 0.8f };
    const int   Ks[3]   = { 32, 64, 128 };
    const int   coff[3] = { 0, 128, 384 };
    for (int br = 0; br < 3; br++) {
      int K = Ks[br]; int R = Bn * S2 * K;
      float r2 = rads[br] * rads[br];
      k_ballquery<<<blk((long)Bn * S2), dim3(TPB), 0, stream>>>(l1x, S1, l2x, S2, r2, K, gi, Bn * S2);
      k_build_feat<<<blk((long)R), dim3(TPB), 0, stream>>>(l1p, 320, 320, l1x, l2x, gi,
                                                           S1, S2, K, 324, bufA, R);
      float* Y; const float* sc; const float* sh;
      run_mlp(stream, sa2[br], 3, bufA, bufB, R, 324, stats, wpad, bpad,
              scale0, shift0, scale1, shift1, &Y, &sc, &sh);
      int O = sa2[br][2].cout;
      k_maxpool<<<blk((long)Bn * S2 * O), dim3(TPB), 0, stream>>>(Y, K, O, sc, sh,
                                                                  l2p, 640, coff[br], Bn * S2 * O);
    }
  }

  // -------- SA3 (group-all on 128 pts -> global 1024-d feature)
  {
    int R = Bn * S2;
    k_feat_all<<<blk((long)R), dim3(TPB), 0, stream>>>(l2p, 640, l2x, 644, bufA, R);
    float* Y; const float* sc; const float* sh;
    run_mlp(stream, sa3, 3, bufA, bufB, R, 644, stats, wpad, bpad,
            scale0, shift0, scale1, shift1, &Y, &sc, &sh);
    k_maxpool<<<blk((long)Bn * 1024), dim3(TPB), 0, stream>>>(Y, S2, 1024, sc, sh,
                                                              gbuf, 1024, 0, Bn * 1024);
  }

  // -------- head: FC(1024->512) BN ReLU, FC(512->256) BN ReLU, FC(256->27)
  {
    // L1
    k_padW<<<blk(512 * 1024), dim3(TPB), 0, stream>>>(fcl[0].W, 512, 512, 1024, 1024,
                                                      wpad, fcl[0].b, bpad, 512 * 1024);
    k_zero<<<dim3(8), dim3(TPB), 0, stream>>>(stats, 2048);
    launch_gemm(stream, gbuf, 1024, wpad, bpad, bufA, 16, 512, nullptr, nullptr,
                stats, stats + 1024);
    k_finalize_bn<<<blk(512), dim3(TPB), 0, stream>>>(stats, stats + 1024, 16.0f,
        fcl[0].g, fcl[0].be, scale0, shift0, 512);
    // L2
    k_padW<<<blk(256 * 512), dim3(TPB), 0, stream>>>(fcl[1].W, 256, 256, 512, 512,
                                                     wpad, fcl[1].b, bpad, 256 * 512);
    k_zero<<<dim3(8), dim3(TPB), 0, stream>>>(stats, 2048);
    launch_gemm(stream, bufA, 512, wpad, bpad, bufB, 16, 256, scale0, shift0,
                stats, stats + 1024);
    k_finalize_bn<<<blk(256), dim3(TPB), 0, stream>>>(stats, stats + 1024, 16.0f,
        fcl[1].g, fcl[1].be, scale1, shift1, 256);
    // L3 (27 outputs, weights padded to 32 rows; no BN)
    k_padW<<<blk(32 * 256), dim3(TPB), 0, stream>>>(W3, 32, 27, 256, 256,
                                                    wpad, b3, bpad, 32 * 256);
    launch_gemm(stream, bufB, 256, wpad, bpad, bufA, 16, 32, scale1, shift1,
                nullptr, nullptr);
    k_copyout<<<blk(16 * 27), dim3(TPB), 0, stream>>>(bufA, (float*)d_out, 16 * 27);
  }
}